// Yolov5_48850958025086
// MI455X (gfx1250) — compile-verified
//
#include <hip/hip_runtime.h>
#include <hip/hip_bf16.h>
#include <stdint.h>

#define BATCH   64
#define NPRED   25200
#define TOPK    1024
#define MAXDET  300
#define CONF_T  0.25f
#define IOU_T   0.45f
#define MAXWH   4096.0f
#define EPSV    1e-7f

typedef __attribute__((ext_vector_type(2))) float v2f;
typedef __attribute__((ext_vector_type(8))) float v8f;

// ---------------------------------------------------------------------------
// K1: decode conf/cand and build distinct 64-bit descending-sort keys.
// key = (monotone_uint(score) << 32) | ~index  -> strict total order matching
// jax.lax.top_k (descending, smaller index wins on ties).
// ---------------------------------------------------------------------------
__global__ __launch_bounds__(256) void yolo_decode_keys(
    const float* __restrict__ pred, unsigned long long* __restrict__ k64) {
  int gid = blockIdx.x * 256 + threadIdx.x;
  if (gid >= BATCH * NPRED) return;
  const float4* row = (const float4*)(pred + (size_t)gid * 8);
  float4 r1 = row[1];                    // obj, c0, c1, c2
  float obj = r1.x;
  float c0 = r1.y * obj, c1 = r1.z * obj, c2 = r1.w * obj;
  float conf = fmaxf(c0, fmaxf(c1, c2));
  float s = ((obj > CONF_T) && (conf > CONF_T)) ? conf : -1.0f;
  unsigned u = __float_as_uint(s);
  unsigned fk = (u & 0x80000000u) ? ~u : (u | 0x80000000u);
  unsigned i = (unsigned)(gid % NPRED);
  k64[gid] = ((unsigned long long)fk << 32) | (unsigned long long)(~i);
}

// ---------------------------------------------------------------------------
// K2: per-image exact top-1024 via 8x8-bit radix select on distinct keys,
// compaction, LDS bitonic sort (descending), then gather box/cls/score.
// ---------------------------------------------------------------------------
__global__ __launch_bounds__(1024) void yolo_topk(
    const float* __restrict__ pred, const unsigned long long* __restrict__ k64,
    float4* __restrict__ bK, int* __restrict__ cK, float* __restrict__ sK) {
  __shared__ unsigned hist[256];
  __shared__ unsigned long long selS[TOPK];
  __shared__ unsigned long long prefixS;
  __shared__ unsigned rS, selCnt;
  const int t = threadIdx.x;
  const int b = blockIdx.x;
  const unsigned long long* keys = k64 + (size_t)b * NPRED;

  if (t == 0) { prefixS = 0ull; rS = TOPK; selCnt = 0u; }
  selS[t] = 0ull;
  __syncthreads();

  for (int p = 7; p >= 0; --p) {
    if (t < 256) hist[t] = 0u;
    __syncthreads();
    unsigned long long prefix = prefixS;
    unsigned long long hiMask = (p == 7) ? 0ull : (~0ull << ((p + 1) * 8));
    for (int e = t; e < NPRED; e += 1024) {
      unsigned long long key = keys[e];
      if ((key & hiMask) == (prefix & hiMask))
        atomicAdd(&hist[(unsigned)(key >> (p * 8)) & 0xFFu], 1u);
    }
    __syncthreads();
    if (t == 0) {
      unsigned r = rS, cum = 0u; int bsel = 0;
      for (int bn = 255; bn >= 0; --bn) {
        unsigned h = hist[bn];
        if (cum + h >= r) { bsel = bn; break; }
        cum += h;
      }
      prefixS = prefix | ((unsigned long long)(unsigned)bsel << (p * 8));
      rS = r - cum;
    }
    __syncthreads();
  }
  unsigned long long T = prefixS;   // 1024th largest key (keys distinct)

  for (int e = t; e < NPRED; e += 1024) {
    unsigned long long key = keys[e];
    if (key >= T) {
      unsigned pos = atomicAdd(&selCnt, 1u);
      if (pos < TOPK) selS[pos] = key;
    }
  }
  __syncthreads();

  // bitonic sort, descending
  for (unsigned k = 2; k <= TOPK; k <<= 1) {
    for (unsigned j = k >> 1; j > 0; j >>= 1) {
      unsigned ixj = (unsigned)t ^ j;
      if (ixj > (unsigned)t) {
        unsigned long long a = selS[t], c = selS[ixj];
        bool desc = (((unsigned)t & k) == 0u);
        if (desc ? (a < c) : (a > c)) { selS[t] = c; selS[ixj] = a; }
      }
      __syncthreads();
    }
  }

  // gather: recover index + score, recompute box (xywh->xyxy) and argmax cls
  unsigned long long key = selS[t];
  unsigned idx = ~((unsigned)key);
  if (idx >= NPRED) idx = NPRED - 1;
  unsigned fk = (unsigned)(key >> 32);
  unsigned u = (fk & 0x80000000u) ? (fk ^ 0x80000000u) : ~fk;
  float s = __uint_as_float(u);
  const float4* row = (const float4*)(pred + ((size_t)b * NPRED + idx) * 8);
  float4 r0 = row[0], r1 = row[1];
  float hx = r0.z * 0.5f, hy = r0.w * 0.5f;
  float obj = r1.x;
  float c0 = r1.y * obj, c1 = r1.z * obj, c2 = r1.w * obj;
  int cls = (c0 >= c1 && c0 >= c2) ? 0 : ((c1 >= c2) ? 1 : 2);
  bK[b * TOPK + t] = make_float4(r0.x - hx, r0.y - hy, r0.x + hx, r0.y + hy);
  cK[b * TOPK + t] = cls;
  sK[b * TOPK + t] = s;
}

// ---------------------------------------------------------------------------
// K3: per-image NMS. 1024 threads (32 wave32s). 176KB LDS (CDNA5 WGP budget).
// Boxes + scores staged global->LDS with async-to-LDS (ASYNCcnt path).
// Full 1024x1024 suppression bitmask built 16x16-tile-wise: WMMA f32
// 16x16x4 computes the pairwise area-sum term of the IoU denominator; the
// 32 pair-decisions of a tile-row-pair are collapsed with a wave32 ballot
// into two ds_or writes (no divergent per-lane atomics).
// Then a single-wave greedy bit-sweep, stable ranking, top-300 emission.
// ---------------------------------------------------------------------------
__global__ __launch_bounds__(1024) void yolo_nms_out(
    const float4* __restrict__ bK, const int* __restrict__ cK,
    const float* __restrict__ sK, const int* __restrict__ im0_h,
    const int* __restrict__ im0_w, float* __restrict__ out) {
  __shared__ float x1s[TOPK], y1s[TOPK], x2s[TOPK], y2s[TOPK];
  __shared__ float areaS[TOPK], sS[TOPK];
  __shared__ float4 boxStage[TOPK];          // 16 KB async staging
  __shared__ unsigned supS[TOPK * 32];       // 128 KB suppression bitmask
  __shared__ int keepS[TOPK];
  __shared__ int scanS[TOPK];
  const int t = threadIdx.x;
  const int b = blockIdx.x;
  const int wave = t >> 5, lane = t & 31;

  // CDNA5 async copies: scores (b32) and boxes (b128) global -> LDS directly
  {
    unsigned ldsa = (unsigned)(uintptr_t)(&sS[t]);
    unsigned long long ga =
        (unsigned long long)(uintptr_t)(sK + (size_t)b * TOPK + t);
    asm volatile("global_load_async_to_lds_b32 %0, %1, off"
                 :: "v"(ldsa), "v"(ga) : "memory");
    unsigned ldsb = (unsigned)(uintptr_t)(&boxStage[t]);
    unsigned long long gb =
        (unsigned long long)(uintptr_t)(bK + (size_t)b * TOPK + t);
    asm volatile("global_load_async_to_lds_b128 %0, %1, off"
                 :: "v"(ldsb), "v"(gb) : "memory");
  }
  __builtin_prefetch(&cK[(size_t)b * TOPK + t], 0, 0);  // global_prefetch_b8

  int cls = cK[(size_t)b * TOPK + t];
  keepS[t] = 0;
  for (int q = t; q < TOPK * 32; q += 1024) supS[q] = 0u;

  asm volatile("s_wait_asynccnt 0x0" ::: "memory");
  float4 bx = boxStage[t];
  float off = (float)cls * MAXWH;            // class-offset boxes for IoU
  x1s[t] = bx.x + off; y1s[t] = bx.y + off;
  x2s[t] = bx.z + off; y2s[t] = bx.w + off;
  areaS[t] = (bx.z - bx.x) * (bx.w - bx.y);  // offset cancels in area
  __syncthreads();

  // Build suppression mask: 64x64 tiles of 16x16 pairs, wave-uniform tiles.
  for (int tile = wave; tile < 64 * 64; tile += 32) {
    int ti = tile >> 6, tj = tile & 63;
    int i0 = ti << 4, j0 = tj << 4;
    if (j0 + 15 <= i0) continue;             // no j>i pair in tile (uniform)
    bool lo = lane < 16;
    int l16 = lane & 15;
    // D[m][n] = area[i0+m] + area[j0+n] via rank-2 outer product:
    // A = [area_i, 1, 0, 0] (16x4), B = [1; area_j; 0; 0] (4x16)
    v2f a, bb;
    a.x  = lo ? areaS[i0 + l16] : 0.0f;      // A: K=0,1 in lanes 0-15
    a.y  = lo ? 1.0f            : 0.0f;
    bb.x = lo ? 1.0f            : 0.0f;      // B: K=0,1 in lanes 0-15
    bb.y = lo ? areaS[j0 + l16] : 0.0f;
    v8f c = {0.f, 0.f, 0.f, 0.f, 0.f, 0.f, 0.f, 0.f};
    v8f d = __builtin_amdgcn_wmma_f32_16x16x4_f32(
        false, a, false, bb, (short)0, c, false, false);

    int j = j0 + l16;                        // this lane's fixed column
    float xj1 = x1s[j], yj1 = y1s[j], xj2 = x2s[j], yj2 = y2s[j];
    const int mbase = lo ? 0 : 8;            // C/D layout: M = v + 8*(lane>=16)
    const int word = j0 >> 5;                // shared mask word for the tile
    const unsigned sh = (unsigned)(j0 & 16); // bit offset within the word

    // One ballot per row-pair: low 16 lanes decide row i0+V, high 16 lanes
    // decide row i0+8+V; two ds_or writes cover all 32 pair decisions.
#define NMS_ROWPAIR(V)                                                        \
    {                                                                         \
      int i = i0 + mbase + (V);                                               \
      float iw  = fminf(x2s[i], xj2) - fmaxf(x1s[i], xj1);                    \
      float ihh = fminf(y2s[i], yj2) - fmaxf(y1s[i], yj1);                    \
      iw = fmaxf(iw, 0.0f); ihh = fmaxf(ihh, 0.0f);                           \
      float inter = iw * ihh;                                                 \
      float iou = inter / (d[(V)] - inter + EPSV);                            \
      bool pred = (j > i) && (iou > IOU_T);                                   \
      unsigned ball = __builtin_amdgcn_ballot_w32(pred);                      \
      if (lane == 0) {                                                        \
        unsigned m = ball & 0xFFFFu;                                          \
        if (m) atomicOr(&supS[((i0 + (V)) << 5) + word], m << sh);            \
      }                                                                       \
      if (lane == 16) {                                                       \
        unsigned m = ball >> 16;                                              \
        if (m) atomicOr(&supS[((i0 + 8 + (V)) << 5) + word], m << sh);        \
      }                                                                       \
    }
    NMS_ROWPAIR(0) NMS_ROWPAIR(1) NMS_ROWPAIR(2) NMS_ROWPAIR(3)
    NMS_ROWPAIR(4) NMS_ROWPAIR(5) NMS_ROWPAIR(6) NMS_ROWPAIR(7)
#undef NMS_ROWPAIR
  }
  __syncthreads();

  // Greedy sweep (exactly the reference fori_loop): single wave, lane l
  // holds removed-bitmask word l in a register; broadcast via readlane.
  if (wave == 0) {
    unsigned remv = 0u;
    for (int i = 0; i < TOPK; ++i) {
      unsigned rm = (unsigned)__builtin_amdgcn_readlane((int)remv, i >> 5);
      bool removed = (rm >> (i & 31)) & 1u;
      bool valid = sS[i] > 0.0f;
      if (valid && !removed) {
        remv |= supS[(i << 5) + lane];
        if (lane == 0) keepS[i] = 1;
      }
    }
  }
  __syncthreads();

  // Stable rank reproducing top_k(where(keep, sK, -1), 300):
  // kept entries (already sorted desc) first, then rejected (-1.0) in index order.
  int k = keepS[t];
  scanS[t] = k;
  __syncthreads();
  for (int o = 1; o < TOPK; o <<= 1) {
    int val = (t >= o) ? scanS[t - o] : 0;
    __syncthreads();
    scanS[t] += val;
    __syncthreads();
  }
  int inc = scanS[t];
  int total = scanS[TOPK - 1];
  int dest = k ? (inc - 1) : (total + (t - inc));

  if (dest < MAXDET) {
    float s = k ? sS[t] : -1.0f;
    float fih = (float)(*im0_h), fiw = (float)(*im0_w);
    float gain = fminf(640.0f / fih, 640.0f / fiw);
    float padx = (640.0f - fiw * gain) * 0.5f;
    float pady = (640.0f - fih * gain) * 0.5f;
    float ox1 = __builtin_rintf(fminf(fmaxf((bx.x - padx) / gain, 0.0f), fiw));
    float oy1 = __builtin_rintf(fminf(fmaxf((bx.y - pady) / gain, 0.0f), fih));
    float ox2 = __builtin_rintf(fminf(fmaxf((bx.z - padx) / gain, 0.0f), fiw));
    float oy2 = __builtin_rintf(fminf(fmaxf((bx.w - pady) / gain, 0.0f), fih));
    float* ob = out + ((size_t)b * MAXDET + dest) * 4;
    ob[0] = ox1; ob[1] = oy1; ob[2] = ox2; ob[3] = oy2;
    const size_t S0 = (size_t)BATCH * MAXDET * 4;
    const size_t S1 = S0 + (size_t)BATCH * MAXDET;
    const size_t S2 = S1 + (size_t)BATCH * MAXDET;
    out[S0 + (size_t)b * MAXDET + dest] = s;
    out[S1 + (size_t)b * MAXDET + dest] = (float)(cls + 1);
    out[S2 + (size_t)b * MAXDET + dest] = (s > 0.0f) ? 1.0f : 0.0f;
  }
}

// ---------------------------------------------------------------------------
extern "C" void kernel_launch(void* const* d_in, const int* in_sizes, int n_in,
                              void* d_out, int out_size, void* d_ws,
                              size_t ws_size, hipStream_t stream) {
  (void)in_sizes; (void)n_in; (void)out_size; (void)ws_size;
  const float* pred = (const float*)d_in[0];
  const int* im0_h = (const int*)d_in[1];
  const int* im0_w = (const int*)d_in[2];

  char* ws = (char*)d_ws;
  unsigned long long* k64 = (unsigned long long*)ws;
  size_t off = (size_t)BATCH * NPRED * sizeof(unsigned long long); // 12.9 MB
  float4* bK = (float4*)(ws + off); off += (size_t)BATCH * TOPK * sizeof(float4);
  int*    cK = (int*)(ws + off);    off += (size_t)BATCH * TOPK * sizeof(int);
  float*  sK = (float*)(ws + off);

  yolo_decode_keys<<<(BATCH * NPRED + 255) / 256, 256, 0, stream>>>(pred, k64);
  yolo_topk<<<BATCH, 1024, 0, stream>>>(pred, k64, bK, cK, sK);
  yolo_nms_out<<<BATCH, 1024, 0, stream>>>(bK, cK, sK, im0_h, im0_w,
                                           (float*)d_out);
}